// GLCMModule_21010980012166
// MI455X (gfx1250) — compile-verified
//
#include <hip/hip_runtime.h>
#include <hip/hip_bf16.h>
#include <math.h>

// ---------------------------------------------------------------------------
// GLCM texture features on MI455X (gfx1250).
//   x: (16, 3, 16, 224, 224) fp32 -> feats (16, 96) fp32
//
// Pipeline:
//   1) quantize_kernel : g = floor(mean_C(x)*255) as u8 (12.8 MB, L2-resident)
//                        + exact per-image std via integer LDS atomics
//   2) glcm_kernel     : one WG per (image, offset); 256 KB LDS histogram
//                        (CDNA5 320 KB LDS/WGP makes this possible);
//                        linear features = sum h*w(i-j)/S (scalar reduction);
//                        ASM = trace(S S^T)/(2S)^2 via V_WMMA_F32_16X16X4_F32
//   3) finalize_kernel : assemble [std, con, dis, hom, asm, sqrt(asm)]
// ---------------------------------------------------------------------------

typedef __attribute__((ext_vector_type(2))) float v2f;
typedef __attribute__((ext_vector_type(8))) float v8f;

#define IMG_N    256          // B*F images
#define HW_PIX   50176        // 224*224
#define HIST_SZ  65536        // 256*256 bins
#define LDS_BYTES (HIST_SZ * 4 + 16)

// ---------------------------------------------------------------- init ------
__global__ void init_kernel(float* __restrict__ accum) {
    int t = threadIdx.x + blockIdx.x * blockDim.x;
    if (t < IMG_N * 4) accum[t] = 0.0f;
}

// ------------------------------------------------------------ quantize ------
// One block per image n = b*16 + f.  g[n, h, w] = floor(mean_c(x)*255).
// Exact integer sum / sumsq for std (values 0..255, fits u32 per-thread).
__global__ void quantize_kernel(const float* __restrict__ x,
                                unsigned char* __restrict__ g,
                                float* __restrict__ stdv) {
    __shared__ unsigned long long ssum, ssumsq;
    const int n = blockIdx.x;
    const int b = n >> 4, f = n & 15;
    const int tid = threadIdx.x;
    if (tid == 0) { ssum = 0ull; ssumsq = 0ull; }
    __syncthreads();

    const float* x0 = x + ((size_t)(b * 3 + 0) * 16 + f) * (size_t)HW_PIX;
    const float* x1 = x + ((size_t)(b * 3 + 1) * 16 + f) * (size_t)HW_PIX;
    const float* x2 = x + ((size_t)(b * 3 + 2) * 16 + f) * (size_t)HW_PIX;
    unsigned char* go = g + (size_t)n * HW_PIX;

    unsigned int psum = 0u, psumsq = 0u;
    for (int p = tid; p < HW_PIX; p += 256) {
        float mean = (x0[p] + x1[p] + x2[p]) / 3.0f;
        int gv = (int)floorf(mean * 255.0f);
        gv = gv < 0 ? 0 : (gv > 255 ? 255 : gv);
        go[p] = (unsigned char)gv;
        psum   += (unsigned int)gv;
        psumsq += (unsigned int)(gv * gv);
    }
    atomicAdd(&ssum,   (unsigned long long)psum);
    atomicAdd(&ssumsq, (unsigned long long)psumsq);
    __syncthreads();

    if (tid == 0) {
        double N    = (double)HW_PIX;
        double mean = (double)ssum / N;
        double var  = (double)ssumsq / N - mean * mean;
        stdv[n] = (float)sqrt(var > 0.0 ? var : 0.0);
    }
}

// ---------------------------------------------------------------- glcm ------
// grid = 1024 blocks: n = blockIdx>>2, offset = blockIdx&3. 1024 threads.
// Dynamic LDS: u32 hist[65536] (256 KB) + float acc[4].
__global__ void glcm_kernel(const unsigned char* __restrict__ g,
                            float* __restrict__ accum) {
    extern __shared__ unsigned char smem[];
    unsigned int* hist = (unsigned int*)smem;
    float* acc = (float*)(smem + (size_t)HIST_SZ * 4);

    const int tid = threadIdx.x;
    const int n   = blockIdx.x >> 2;
    const int off = blockIdx.x & 3;

    const int DR[4] = {0, 1, 1, 1};
    const int DC[4] = {1, 1, 0, -1};
    const int dr = DR[off], dc = DC[off];
    const int R  = 224 - dr;
    const int C  = 224 - (dc < 0 ? -dc : dc);
    const int aoff = (dc < 0) ? 1 : 0;   // a column shift
    const int boff = (dc > 0) ? 1 : 0;   // b column shift

    // zero histogram + accumulators (bank-conflict-free interleave)
    #pragma unroll
    for (int q = 0; q < HIST_SZ / 1024; ++q) hist[q * 1024 + tid] = 0u;
    if (tid < 4) acc[tid] = 0.0f;
    __syncthreads();

    // ---- build co-occurrence histogram with LDS atomics ----
    const unsigned char* gi = g + (size_t)n * HW_PIX;
    const int total = R * C;
    for (int t = tid; t < total; t += 1024) {
        int r = t / C;
        int c = t - r * C;
        __builtin_prefetch(gi + (r + dr) * 224 + c + boff + 2048, 0, 0);
        int a = gi[r * 224 + c + aoff];
        int b = gi[(r + dr) * 224 + c + boff];
        atomicAdd(&hist[(a << 8) | b], 1u);
    }
    __syncthreads();

    // ---- linear features: sum_bins h * w(i-j)  (w symmetric => no H^T) ----
    float pc = 0.0f, pd = 0.0f, ph = 0.0f;
    for (int q = 0; q < HIST_SZ / 1024; ++q) {
        int idx = q * 1024 + tid;
        unsigned int h = hist[idx];
        if (h) {
            float d  = (float)((idx >> 8) - (idx & 255));
            float d2 = d * d;
            float hf = (float)h;
            pc += hf * d2;
            pd += hf * fabsf(d);
            ph += hf / (1.0f + d2);
        }
    }
    atomicAdd(&acc[0], pc);
    atomicAdd(&acc[1], pd);
    atomicAdd(&acc[2], ph);

    // ---- ASM = trace(S S^T) via V_WMMA_F32_16X16X4_F32, S = H + H^T ----
    // 32 waves; wave w owns diagonal block mb = w>>1 and K-half (w&1)*128.
    // A[m][k] = S[mb*16+m][k] lives in lane (k>=2?16:0)+m, vgpr k%2.
    // B[k][n] = S^T[k][mb*16+n] = S[mb*16+n][k] -> identical per-lane values,
    // so one v2f feeds both operands (D = A * A^T over this K chunk).
    {
        const int wave  = tid >> 5;
        const int lane  = tid & 31;
        const int mb    = wave >> 1;
        const int kbase = (wave & 1) << 7;
        const int m     = lane & 15;
        const int kh    = lane >> 4;           // 0 or 1
        const int row   = mb * 16 + m;
        v8f cacc = {};
        for (int s = 0; s < 32; ++s) {
            int k0 = kbase + (s << 2) + (kh << 1);
            float sx = (float)(hist[(row << 8) + k0]     + hist[( k0      << 8) + row]);
            float sy = (float)(hist[(row << 8) + k0 + 1] + hist[((k0 + 1) << 8) + row]);
            v2f ab;
            ab.x = sx;
            ab.y = sy;
            cacc = __builtin_amdgcn_wmma_f32_16x16x4_f32(
                false, ab, false, ab, (short)0, cacc, false, false);
        }
        // Diagonal of 16x16 f32 D: D[r][r] in lane r (vgpr r) for r<8,
        // D[8+r][8+r] in lane 24+r (vgpr r).
        const bool lo = lane < 8, hi = lane >= 24;
        const int  rsel = lo ? lane : (lane - 24);
        float val = 0.0f;
        #pragma unroll
        for (int q = 0; q < 8; ++q) {
            float cq = cacc[q];
            val = ((lo || hi) && rsel == q) ? cq : val;
        }
        atomicAdd(&acc[3], val);
    }
    __syncthreads();

    // ---- per-offset normalization, averaged over 4 offsets ----
    if (tid == 0) {
        const float Sp  = (float)total;          // pairs for this offset
        const float inv = 0.25f / Sp;            // /S, avg over 4 offsets
        atomicAdd(&accum[n * 4 + 0], acc[0] * inv);
        atomicAdd(&accum[n * 4 + 1], acc[1] * inv);
        atomicAdd(&accum[n * 4 + 2], acc[2] * inv);
        atomicAdd(&accum[n * 4 + 3], acc[3] * (0.25f / (4.0f * Sp * Sp)));
    }
}

// ------------------------------------------------------------ finalize ------
__global__ void finalize_kernel(const float* __restrict__ accum,
                                const float* __restrict__ stdv,
                                float* __restrict__ out) {
    int n = threadIdx.x;
    if (n < IMG_N) {
        int b = n >> 4, f = n & 15;
        float con = accum[n * 4 + 0];
        float dis = accum[n * 4 + 1];
        float hom = accum[n * 4 + 2];
        float as  = accum[n * 4 + 3];
        float* o = out + b * 96 + f * 6;
        o[0] = stdv[n];
        o[1] = con;
        o[2] = dis;
        o[3] = hom;
        o[4] = as;
        o[5] = sqrtf(as);
    }
}

// -------------------------------------------------------------- launch ------
extern "C" void kernel_launch(void* const* d_in, const int* in_sizes, int n_in,
                              void* d_out, int out_size, void* d_ws, size_t ws_size,
                              hipStream_t stream) {
    (void)in_sizes; (void)n_in; (void)out_size; (void)ws_size;
    const float* x = (const float*)d_in[0];
    float* out = (float*)d_out;

    // d_ws layout: [u8 g: 12,845,056 B][f32 accum: 256*4][f32 std: 256]
    unsigned char* g = (unsigned char*)d_ws;
    const size_t G_BYTES = (size_t)IMG_N * HW_PIX;
    float* accum = (float*)((char*)d_ws + G_BYTES);
    float* stdv  = accum + IMG_N * 4;

    // allow 256 KB dynamic LDS for the histogram kernel
    hipFuncSetAttribute((const void*)glcm_kernel,
                        hipFuncAttributeMaxDynamicSharedMemorySize, LDS_BYTES);

    init_kernel<<<1, 1024, 0, stream>>>(accum);
    quantize_kernel<<<IMG_N, 256, 0, stream>>>(x, g, stdv);
    glcm_kernel<<<IMG_N * 4, 1024, LDS_BYTES, stream>>>(g, accum);
    finalize_kernel<<<1, 256, 0, stream>>>(accum, stdv, out);
}